// ChipPairEncoderAlpha_76682346102982
// MI455X (gfx1250) — compile-verified
//
#include <hip/hip_runtime.h>

// Sizes from the reference: B=8, N=128, CM=32, H=4, CP=16, all float32.
#define BB 8
#define NSEQ 128
#define CMC 32
#define NH 4
#define CPC 16

typedef float v2f __attribute__((ext_vector_type(2)));
typedef float v8f __attribute__((ext_vector_type(8)));

static __device__ __forceinline__ float wave_sum32(float x) {
#pragma unroll
  for (int m = 1; m < 32; m <<= 1) x += __shfl_xor(x, m, 32);
  return x;
}
static __device__ __forceinline__ float half_sum16(float x) {
#pragma unroll
  for (int m = 1; m < 16; m <<= 1) x += __shfl_xor(x, m, 32);
  return x;
}
static __device__ __forceinline__ float half_max16(float x) {
#pragma unroll
  for (int m = 1; m < 16; m <<= 1) x = fmaxf(x, __shfl_xor(x, m, 32));
  return x;
}

// CDNA5 f32 WMMA: D(16x16) = A(16x4) * B(4x16) + C, fp32 in/out.
static __device__ __forceinline__ v8f wmma4(v2f a, v2f b, v8f c) {
  return __builtin_amdgcn_wmma_f32_16x16x4_f32(false, a, false, b, (short)0, c,
                                               false, false);
}

// ---------------------------------------------------------------------------
// 1) Embed: chip = max_t LN(sig_t * w + b + pos); also per-row L2 norms.
//    One wave per (side, b, n); lane = channel.
// ---------------------------------------------------------------------------
__global__ void k_embed(const float* __restrict__ t0r, const float* __restrict__ t1r,
                        const float* __restrict__ t2r, const float* __restrict__ t3r,
                        const float* __restrict__ t0c, const float* __restrict__ t1c,
                        const float* __restrict__ t2c, const float* __restrict__ t3c,
                        const float* __restrict__ w, const float* __restrict__ bm,
                        const float* __restrict__ posr, const float* __restrict__ posc,
                        const float* __restrict__ g, const float* __restrict__ be,
                        float* __restrict__ chipI, float* __restrict__ chipJ,
                        float* __restrict__ normI, float* __restrict__ normJ) {
  int idx = blockIdx.x;
  int lane = threadIdx.x;
  int side = (idx >= BB * NSEQ) ? 1 : 0;
  int bn = side ? (idx - BB * NSEQ) : idx;
  int n = bn & (NSEQ - 1);
  const float* tk0 = side ? t0c : t0r;
  const float* tk1 = side ? t1c : t1r;
  const float* tk2 = side ? t2c : t2r;
  const float* tk3 = side ? t3c : t3r;
  const float* pos = side ? posc : posr;
  const float* tracks[4] = {tk0, tk1, tk2, tk3};
  float wc = w[lane], bc = bm[lane], pc = pos[n * CMC + lane];
  float gc = g[lane], bec = be[lane];
  float acc = -3.0e38f;
#pragma unroll
  for (int t = 0; t < 4; ++t) {
    float val = tracks[t][bn] * wc + bc + pc;
    float mean = wave_sum32(val) * (1.0f / 32.0f);
    float dv = val - mean;
    float var = wave_sum32(dv * dv) * (1.0f / 32.0f);
    float ln = dv * rsqrtf(var + 1e-5f) * gc + bec;
    acc = fmaxf(acc, ln);
  }
  float* chip = side ? chipJ : chipI;
  chip[(size_t)bn * CMC + lane] = acc;
  float nrm = sqrtf(wave_sum32(acc * acc));
  if (lane == 0) (side ? normJ : normI)[bn] = nrm;
}

// ---------------------------------------------------------------------------
// 2) Rearrange W_outer (o, c*32+d) -> Wt[(o*32+d), c] so the T-GEMM is NT.
// ---------------------------------------------------------------------------
__global__ void k_wt(const float* __restrict__ Wouter, float* __restrict__ Wt) {
  int tid = blockIdx.x * blockDim.x + threadIdx.x;  // 32768
  int c = tid & 31;
  int nidx = tid >> 5;
  int o = nidx >> 5;
  int d = nidx & 31;
  Wt[tid] = Wouter[o * 1024 + c * 32 + d];
}

// ---------------------------------------------------------------------------
// 3) Generic WMMA GEMM (NT): D[m,n] = sum_k A[m,k]*W[n,k], K=32 fixed.
//    One wave per 16x16 tile.
// ---------------------------------------------------------------------------
__global__ void k_gemm_nt(const float* __restrict__ A, int lda,
                          const float* __restrict__ W, int ldw,
                          float* __restrict__ D, int ldd, int Mtiles) {
  int tile = blockIdx.x;
  int mt = tile % Mtiles;
  int nt = tile / Mtiles;
  int lane = threadIdx.x;
  int half = lane >> 4;
  int l = lane & 15;
  int m0 = mt * 16, n0 = nt * 16;
  const float* Ar = A + (size_t)(m0 + l) * lda;
  const float* Wr = W + (size_t)(n0 + l) * ldw;
  v8f c = {};
#pragma unroll
  for (int kk = 0; kk < 8; ++kk) {
    int k = kk * 4 + half * 2;
    v2f a, b;
    a[0] = Ar[k];
    a[1] = Ar[k + 1];
    b[0] = Wr[k];
    b[1] = Wr[k + 1];
    c = wmma4(a, b, c);
  }
#pragma unroll
  for (int v = 0; v < 8; ++v) {
    int m = m0 + v + half * 8;
    D[(size_t)m * ldd + n0 + l] = c[v];
  }
}

// ---------------------------------------------------------------------------
// 4) pair[b,i,j,o] = (T[b,i,o,:] . chipJ[b,j,:]) / max(|u_i||v_j|,1e-6) + b_o
//    Batched WMMA GEMM: per b, M=(i,o)=4096, N=j=128, K=32, custom store.
// ---------------------------------------------------------------------------
__global__ void k_gemm_pair(const float* __restrict__ T, const float* __restrict__ chipJ,
                            const float* __restrict__ normI, const float* __restrict__ normJ,
                            const float* __restrict__ bouter, float* __restrict__ pair) {
  int b = blockIdx.z;
  int m0 = blockIdx.x * 16;  // over i*32+o (4096)
  int n0 = blockIdx.y * 16;  // over j (128)
  int lane = threadIdx.x;
  int half = lane >> 4;
  int l = lane & 15;
  const float* Ar = T + ((size_t)b * 4096 + m0 + l) * 32;
  const float* Wr = chipJ + ((size_t)b * NSEQ + n0 + l) * 32;
  v8f c = {};
#pragma unroll
  for (int kk = 0; kk < 8; ++kk) {
    int k = kk * 4 + half * 2;
    v2f a, bq;
    a[0] = Ar[k];
    a[1] = Ar[k + 1];
    bq[0] = Wr[k];
    bq[1] = Wr[k + 1];
    c = wmma4(a, bq, c);
  }
  int j = n0 + l;
  float nj = normJ[b * NSEQ + j];
#pragma unroll
  for (int v = 0; v < 8; ++v) {
    int m = m0 + v + half * 8;
    int i = m >> 5, o = m & 31;
    float sc = 1.0f / fmaxf(normI[b * NSEQ + i] * nj, 1e-6f);
    pair[(((size_t)b * NSEQ + i) * NSEQ + j) * CMC + o] = c[v] * sc + bouter[o];
  }
}

// ---------------------------------------------------------------------------
// 5) Attention core: one block per (b, r, h), 8 waves of 32.
//    x tile staged into LDS with CDNA5 async global->LDS copies (ASYNCcnt);
//    k/v projected into LDS; q built per-wave in WMMA A layout; scores via
//    V_WMMA_F32_16X16X4_F32 (K=8 -> 2 steps); exact softmax in registers;
//    P@V with lane partials + 16-lane shuffle reduce. Writes O (B,N,N,32).
// ---------------------------------------------------------------------------
__global__ void k_attn(const float* __restrict__ X, const float* __restrict__ Wqkv,
                       const float* __restrict__ bulk, const float* __restrict__ wbias,
                       const float* __restrict__ bbias, float* __restrict__ Oout,
                       int transBias) {
  __shared__ float xs[NSEQ * CMC];  // 16 KB
  __shared__ float ks[NSEQ * 8];    // 4 KB
  __shared__ float vs[NSEQ * 8];    // 4 KB
  int blk = blockIdx.x;
  int h = blk & (NH - 1);
  int r = (blk >> 2) & (NSEQ - 1);
  int b = blk >> 9;
  const float* xrow = X + ((size_t)b * NSEQ + r) * NSEQ * CMC;
  int tid = threadIdx.x;

  // Async global -> LDS staging of the 16KB x tile: 16B per lane, 4 rounds.
  // LDS aperture truncates generic addresses to addr[31:0] (ISA 10.2), so the
  // low 32 bits of the generic pointer are the wave-relative LDS byte offset.
  {
    unsigned lds_base = (unsigned)(uintptr_t)(&xs[0]);
#pragma unroll
    for (int it = 0; it < 4; ++it) {
      unsigned lds_addr = lds_base + (unsigned)(tid + it * 256) * 16u;
      const char* gaddr = (const char*)xrow + (size_t)(tid + it * 256) * 16u;
      asm volatile("global_load_async_to_lds_b128 %0, %1, off"
                   :
                   : "v"(lds_addr), "v"(gaddr)
                   : "memory");
    }
    asm volatile("s_wait_asynccnt 0x0" ::: "memory");
  }
  __syncthreads();

  for (int i = tid; i < NSEQ * 8; i += 256) {
    int n = i >> 3, d = i & 7;
    const float* wk = Wqkv + (32 + h * 8 + d) * 32;
    const float* wv = Wqkv + (64 + h * 8 + d) * 32;
    float sk = 0.f, sv = 0.f;
#pragma unroll
    for (int c = 0; c < 32; ++c) {
      float x = xs[n * 32 + c];
      sk += x * wk[c];
      sv += x * wv[c];
    }
    ks[i] = sk;
    vs[i] = sv;
  }
  __syncthreads();

  int lane = tid & 31;
  int wave = tid >> 5;
  int half = lane >> 4;
  int l = lane & 15;
  int m0 = wave * 16;

  // q in WMMA-A layout: qa[kc] covers K = kc*4 .. kc*4+3 (split by lane half)
  v2f qa[2];
#pragma unroll
  for (int kc = 0; kc < 2; ++kc) {
    int d0 = kc * 4 + half * 2;
    const float* w0 = Wqkv + (h * 8 + d0) * 32;
    const float* w1 = Wqkv + (h * 8 + d0 + 1) * 32;
    const float* xr = xs + (m0 + l) * 32;
    float s0 = 0.f, s1 = 0.f;
#pragma unroll
    for (int c = 0; c < 32; ++c) {
      float x = xr[c];
      s0 += x * w0[c];
      s1 += x * w1[c];
    }
    qa[kc][0] = s0;
    qa[kc][1] = s1;
  }

  float wb = wbias[h], bbv = bbias[h];
  const float* bulkB = bulk + (size_t)b * NSEQ * NSEQ;
  const float scale = 0.35355339059327373f;  // 1/sqrt(8)

  v8f s[8];
#pragma unroll
  for (int t = 0; t < 8; ++t) {
    v8f acc = {};
#pragma unroll
    for (int kc = 0; kc < 2; ++kc) {
      int d0 = kc * 4 + half * 2;
      v2f bq;
      bq[0] = ks[(t * 16 + l) * 8 + d0];
      bq[1] = ks[(t * 16 + l) * 8 + d0 + 1];
      acc = wmma4(qa[kc], bq, acc);
    }
    int j = t * 16 + l;
#pragma unroll
    for (int v = 0; v < 8; ++v) {
      int m = m0 + v + half * 8;
      float bval = transBias ? bulkB[j * NSEQ + m] : bulkB[m * NSEQ + j];
      acc[v] = acc[v] * scale + bval * wb + bbv;
    }
    s[t] = acc;
  }

  // softmax over keys (cols split across 16 lanes and 8 tiles)
  float rsum[8];
#pragma unroll
  for (int v = 0; v < 8; ++v) {
    float mx = -3.0e38f;
#pragma unroll
    for (int t = 0; t < 8; ++t) mx = fmaxf(mx, s[t][v]);
    mx = half_max16(mx);
    float sum = 0.f;
#pragma unroll
    for (int t = 0; t < 8; ++t) {
      float p = __expf(s[t][v] - mx);
      s[t][v] = p;
      sum += p;
    }
    rsum[v] = half_sum16(sum);
  }

  // O = P @ V (16 x 8 per wave), then store
  float* Orow = Oout + ((size_t)b * NSEQ + r) * NSEQ * CMC;
#pragma unroll
  for (int v = 0; v < 8; ++v) {
    int m = m0 + v + half * 8;
    float inv = 1.0f / rsum[v];
#pragma unroll
    for (int d = 0; d < 8; ++d) {
      float acc = 0.f;
#pragma unroll
      for (int t = 0; t < 8; ++t) acc += s[t][v] * vs[(t * 16 + l) * 8 + d];
      acc = half_sum16(acc);
      if (l == d) Orow[m * CMC + h * 8 + d] = acc * inv;
    }
  }
}

// ---------------------------------------------------------------------------
// 6) x = LN(x + O @ Wout^T); wave-per-row, lane = channel. In place.
// ---------------------------------------------------------------------------
__global__ void k_outproj_ln(float* __restrict__ X, const float* __restrict__ O,
                             const float* __restrict__ Wout, const float* __restrict__ g,
                             const float* __restrict__ be) {
  int row = blockIdx.x * 8 + (threadIdx.x >> 5);
  int lane = threadIdx.x & 31;
  float* xr = X + (size_t)row * CMC;
  float oc = O[(size_t)row * CMC + lane];
  const float* wr = Wout + lane * 32;
  float t = 0.f;
#pragma unroll
  for (int c = 0; c < 32; ++c) t += __shfl(oc, c, 32) * wr[c];
  float val = xr[lane] + t;
  float mean = wave_sum32(val) * (1.0f / 32.0f);
  float dv = val - mean;
  float var = wave_sum32(dv * dv) * (1.0f / 32.0f);
  xr[lane] = dv * rsqrtf(var + 1e-5f) * g[lane] + be[lane];
}

// ---------------------------------------------------------------------------
// 7) Transpose (B,N,N,C) dims 1<->2. Wave-per-(b,i,j), lane = channel.
// ---------------------------------------------------------------------------
__global__ void k_transpose(const float* __restrict__ in, float* __restrict__ out) {
  int row = blockIdx.x * 8 + (threadIdx.x >> 5);
  int lane = threadIdx.x & 31;
  int j = row & (NSEQ - 1);
  int i = (row >> 7) & (NSEQ - 1);
  int b = row >> 14;
  out[(((size_t)b * NSEQ + j) * NSEQ + i) * CMC + lane] = in[(size_t)row * CMC + lane];
}

// ---------------------------------------------------------------------------
// 8) Final: y = pair @ Wp^T + bp; y += alpha*LN16(y); y *= sigmoid(y);
//    stored transposed (B,CP,N,N). Wave-per-(b,i,j).
// ---------------------------------------------------------------------------
__global__ void k_final(const float* __restrict__ X, const float* __restrict__ Wp,
                        const float* __restrict__ bp, const float* __restrict__ g,
                        const float* __restrict__ be, const float* __restrict__ alpha,
                        float* __restrict__ out) {
  int row = blockIdx.x * 8 + (threadIdx.x >> 5);
  int lane = threadIdx.x & 31;
  int j = row & (NSEQ - 1);
  int i = (row >> 7) & (NSEQ - 1);
  int b = row >> 14;
  float xc = X[(size_t)row * CMC + lane];
  int p = lane & 15;
  const float* wr = Wp + p * 32;
  float y = 0.f;
#pragma unroll
  for (int c = 0; c < 32; ++c) y += __shfl(xc, c, 32) * wr[c];
  y += bp[p];
  // LN over 16 channels (both halves hold identical copies)
  float mean = half_sum16(y) * (1.0f / 16.0f);
  float dv = y - mean;
  float var = half_sum16(dv * dv) * (1.0f / 16.0f);
  float yl = dv * rsqrtf(var + 1e-5f) * g[p] + be[p];
  float y2 = y + alpha[0] * yl;
  float sw = y2 / (1.0f + __expf(-y2));
  if (lane < 16) out[(((size_t)b * CPC + p) * NSEQ + i) * NSEQ + j] = sw;
}

// ---------------------------------------------------------------------------
extern "C" void kernel_launch(void* const* d_in, const int* in_sizes, int n_in,
                              void* d_out, int out_size, void* d_ws, size_t ws_size,
                              hipStream_t stream) {
  const float* t0r = (const float*)d_in[0];
  const float* t1r = (const float*)d_in[1];
  const float* t2r = (const float*)d_in[2];
  const float* t3r = (const float*)d_in[3];
  const float* t0c = (const float*)d_in[4];
  const float* t1c = (const float*)d_in[5];
  const float* t2c = (const float*)d_in[6];
  const float* t3c = (const float*)d_in[7];
  const float* bulk = (const float*)d_in[8];  // (B,1,N,N) == (B,N,N)
  const float* w_msa = (const float*)d_in[9];
  const float* b_msa = (const float*)d_in[10];
  const float* pos_row = (const float*)d_in[11];
  const float* pos_col = (const float*)d_in[12];
  const float* g_msa = (const float*)d_in[13];
  const float* beta_msa = (const float*)d_in[14];
  const float* W_outer = (const float*)d_in[15];
  const float* b_outer = (const float*)d_in[16];
  const float* W_row_qkv = (const float*)d_in[17];
  const float* W_row_out = (const float*)d_in[18];
  const float* g_row = (const float*)d_in[19];
  const float* beta_row = (const float*)d_in[20];
  const float* w_row_bias = (const float*)d_in[21];
  const float* b_row_bias = (const float*)d_in[22];
  const float* W_col_qkv = (const float*)d_in[23];
  const float* W_col_out = (const float*)d_in[24];
  const float* g_col = (const float*)d_in[25];
  const float* beta_col = (const float*)d_in[26];
  const float* w_col_bias = (const float*)d_in[27];
  const float* b_col_bias = (const float*)d_in[28];
  const float* W_pair = (const float*)d_in[29];
  const float* b_pair = (const float*)d_in[30];
  const float* g_ada = (const float*)d_in[31];
  const float* beta_ada = (const float*)d_in[32];
  const float* alpha = (const float*)d_in[33];
  float* out = (float*)d_out;

  // Workspace layout (floats): ~55 MB total
  float* ws = (float*)d_ws;
  float* chipI = ws;             // 32768
  float* chipJ = chipI + 32768;  // 32768
  float* normI = chipJ + 32768;  // 1024
  float* normJ = normI + 1024;   // 1024
  float* Wt = normJ + 1024;      // 32768
  float* T = Wt + 32768;         // 1048576
  float* pair = T + 1048576;     // 4194304
  float* attO = pair + 4194304;  // 4194304
  float* pairT = attO + 4194304; // 4194304

  // 1) embed both sides
  k_embed<<<2 * BB * NSEQ, 32, 0, stream>>>(t0r, t1r, t2r, t3r, t0c, t1c, t2c, t3c,
                                            w_msa, b_msa, pos_row, pos_col, g_msa,
                                            beta_msa, chipI, chipJ, normI, normJ);
  // 2) rearrange W_outer
  k_wt<<<128, 256, 0, stream>>>(W_outer, Wt);
  // 3) T = chipI @ Wt^T : M=1024, N=1024, K=32
  k_gemm_nt<<<64 * 64, 32, 0, stream>>>(chipI, 32, Wt, 32, T, 1024, 64);
  // 4) pair (with norm scale + b_outer)
  k_gemm_pair<<<dim3(256, 8, BB), 32, 0, stream>>>(T, chipJ, normI, normJ, b_outer, pair);

  // 5) row attention
  k_attn<<<BB * NSEQ * NH, 256, 0, stream>>>(pair, W_row_qkv, bulk, w_row_bias,
                                             b_row_bias, attO, 0);
  k_outproj_ln<<<BB * NSEQ * NSEQ / 8, 256, 0, stream>>>(pair, attO, W_row_out, g_row,
                                                         beta_row);
  // 6) transpose -> column-major view
  k_transpose<<<BB * NSEQ * NSEQ / 8, 256, 0, stream>>>(pair, pairT);
  // 7) column attention (bias uses bulk^T)
  k_attn<<<BB * NSEQ * NH, 256, 0, stream>>>(pairT, W_col_qkv, bulk, w_col_bias,
                                             b_col_bias, attO, 1);
  k_outproj_ln<<<BB * NSEQ * NSEQ / 8, 256, 0, stream>>>(pairT, attO, W_col_out, g_col,
                                                         beta_col);
  // 8) transpose back
  k_transpose<<<BB * NSEQ * NSEQ / 8, 256, 0, stream>>>(pairT, pair);
  // 9) final projection + adaLN + SiLU + output transpose
  k_final<<<BB * NSEQ * NSEQ / 8, 256, 0, stream>>>(pair, W_pair, b_pair, g_ada,
                                                    beta_ada, alpha, out);
}